// ImageRadiomicsSimilarityModule_10574209483557
// MI455X (gfx1250) — compile-verified
//
#include <hip/hip_runtime.h>
#include <hip/hip_bf16.h>
#include <math.h>

typedef __attribute__((ext_vector_type(2))) float v2f;
typedef __attribute__((ext_vector_type(8))) float v8f;

// ---------------------------------------------------------------------------
// Kernel 1: image mean over H*W (256 contiguous floats per (b,c) row).
// One wave32 per row; each lane loads 2x float4 (fully coalesced 128B/cycle),
// then a 5-step shuffle reduction. 134 MB streamed -> this is the HBM-bound hot loop.
// ---------------------------------------------------------------------------
__global__ __launch_bounds__(256) void img_mean_kernel(const float* __restrict__ in,
                                                       float* __restrict__ out) {
    int row  = (blockIdx.x << 3) + (threadIdx.x >> 5);   // (b*512 + c), 131072 rows
    int lane = threadIdx.x & 31;
    const float4* p = reinterpret_cast<const float4*>(in + (size_t)row * 256);
    float4 v0 = p[lane];
    float4 v1 = p[lane + 32];
    float s = (v0.x + v0.y) + (v0.z + v0.w) + (v1.x + v1.y) + (v1.z + v1.w);
    #pragma unroll
    for (int off = 16; off > 0; off >>= 1)
        s += __shfl_xor(s, off, 32);
    if (lane == 0)
        out[row] = s * (1.0f / 256.0f);
}

// ---------------------------------------------------------------------------
// Kernel 2: radiomics mean over axis 1 ([256,32,512] -> [256,512]).
// One block per batch; 512 threads stride over the 32 rows (coalesced).
// ---------------------------------------------------------------------------
__global__ __launch_bounds__(512) void rad_mean_kernel(const float* __restrict__ in,
                                                       float* __restrict__ out) {
    int b = blockIdx.x;
    int t = threadIdx.x;
    const float* p = in + (size_t)b * 32 * 512 + t;
    float s = 0.0f;
    #pragma unroll
    for (int r = 0; r < 32; ++r)
        s += p[r * 512];
    out[b * 512 + t] = s * (1.0f / 32.0f);
}

// ---------------------------------------------------------------------------
// Kernel 3: inverse L2 norms for all 512 embedding rows (256 img + 256 rad).
// invn[row] = 1 / max(||x||, 1e-12)  (matches F.normalize semantics)
// ---------------------------------------------------------------------------
__global__ __launch_bounds__(256) void inv_norm_kernel(const float* __restrict__ img,
                                                       const float* __restrict__ rad,
                                                       float* __restrict__ invn) {
    __shared__ float sm[256];
    int row = blockIdx.x;
    const float* src = (row < 256) ? (img + (size_t)row * 512)
                                   : (rad + (size_t)(row - 256) * 512);
    int t = threadIdx.x;
    float a = src[t];
    float b = src[t + 256];
    sm[t] = a * a + b * b;
    __syncthreads();
    #pragma unroll
    for (int s = 128; s > 0; s >>= 1) {
        if (t < s) sm[t] += sm[t + s];
        __syncthreads();
    }
    if (t == 0)
        invn[row] = 1.0f / fmaxf(sqrtf(sm[0]), 1e-12f);
}

// ---------------------------------------------------------------------------
// Kernel 4: similarity GEMM via V_WMMA_F32_16X16X4_F32.
// One wave per 16x16 output tile (grid = 256 tiles), K = 512 in steps of 4.
// Row scales (1/norm) folded into the A/B operands; 1/temperature on store.
//
// ISA layouts (CDNA5 ISA):
//   A 16x4 f32 : lane L holds M=(L&15), VGPR0/1 = K + 2*(L>>4), +1
//   B 4x16 f32 : lane L holds N=(L&15), VGPR0/1 = K + 2*(L>>4), +1  (B = rad^T)
//   C/D 16x16  : VGPR v -> row v + 8*(L>>4), col (L&15)
// ---------------------------------------------------------------------------
__global__ __launch_bounds__(32) void sim_wmma_kernel(const float* __restrict__ img,
                                                      const float* __restrict__ rad,
                                                      const float* __restrict__ invn,
                                                      const float* __restrict__ temperature,
                                                      float* __restrict__ out) {
    int tile = blockIdx.x;          // 256 tiles: 16 x 16 grid over [256,256]
    int ti = tile >> 4;             // output row tile
    int tj = tile & 15;             // output col tile
    int lane  = threadIdx.x & 31;
    int lhalf = lane >> 4;          // 0: K,K+1   1: K+2,K+3
    int lmod  = lane & 15;

    int rowA = ti * 16 + lmod;      // image row this lane feeds
    int rowB = tj * 16 + lmod;      // radiomics row this lane feeds
    float sA = invn[rowA];
    float sB = invn[256 + rowB];

    const float* pa = img + (size_t)rowA * 512 + 2 * lhalf;
    const float* pb = rad + (size_t)rowB * 512 + 2 * lhalf;

    v8f acc = {};
    #pragma unroll 8
    for (int k = 0; k < 512; k += 4) {
        v2f a = *reinterpret_cast<const v2f*>(pa + k);  // 8B aligned (even element)
        v2f b = *reinterpret_cast<const v2f*>(pb + k);
        a = a * sA;
        b = b * sB;
        acc = __builtin_amdgcn_wmma_f32_16x16x4_f32(
            /*neg_a=*/false, a, /*neg_b=*/false, b,
            /*c_mod=*/(short)0, acc, /*reuse_a=*/false, /*reuse_b=*/false);
    }

    float invt = 1.0f / temperature[0];
    int col = tj * 16 + lmod;
    #pragma unroll
    for (int v = 0; v < 8; ++v) {
        int row = ti * 16 + v + 8 * lhalf;
        out[row * 256 + col] = acc[v] * invt;
    }
}

// ---------------------------------------------------------------------------
// Host-side launch. Outputs concatenated: [sim 256*256][img_emb 256*512][rad_emb 256*512]
// ---------------------------------------------------------------------------
extern "C" void kernel_launch(void* const* d_in, const int* in_sizes, int n_in,
                              void* d_out, int out_size, void* d_ws, size_t ws_size,
                              hipStream_t stream) {
    const float* img_feat = (const float*)d_in[0];   // [256,512,16,16]
    const float* rad_feat = (const float*)d_in[1];   // [256,32,512]
    const float* temp     = (const float*)d_in[2];   // [1]

    float* out_sim = (float*)d_out;                  // [256,256]
    float* out_img = out_sim + 256 * 256;            // [256,512]
    float* out_rad = out_img + 256 * 512;            // [256,512]

    float* invn = (float*)d_ws;                      // [512] inverse norms

    // 131072 rows / 8 waves-per-block
    img_mean_kernel<<<16384, 256, 0, stream>>>(img_feat, out_img);
    rad_mean_kernel<<<256, 512, 0, stream>>>(rad_feat, out_rad);
    inv_norm_kernel<<<512, 256, 0, stream>>>(out_img, out_rad, invn);
    sim_wmma_kernel<<<256, 32, 0, stream>>>(out_img, out_rad, invn, temp, out_sim);
}